// ClasswiseAdaptiveECELoss_2396591751308
// MI455X (gfx1250) — compile-verified
//
#include <hip/hip_runtime.h>
#include <hip/hip_bf16.h>

#ifndef __has_builtin
#define __has_builtin(x) 0
#endif

// Problem constants (from the reference)
#define CB 4
#define CC 19
#define CH 512
#define CW 1024
#define CHW (CH * CW)             // 524288
#define CN (CB * CH * CW)         // 2097152
#define NBINS 15                  // adaptive ECE bins
#define NEDGES (NBINS + 1)        // 16
#define HBINS 512                 // fine histogram bins per class (sqrt-mapped)
#define TPB 256                   // 8 wave32 per block
#define NBLK 256                  // pixels per block = CN/NBLK = 8192 (stays within one b)
#define PIX_PER_BLK (CN / NBLK)   // 8192
#define TILES (PIX_PER_BLK / TPB) // 32

static_assert(CHW % PIX_PER_BLK == 0, "block chunk must stay inside one batch image");
static_assert(PIX_PER_BLK % TPB == 0, "tiles must divide");

// ---------------- CDNA5 async global->LDS staging ----------------
#if __has_builtin(__builtin_amdgcn_global_load_async_to_lds_b32) && \
    __has_builtin(__builtin_amdgcn_s_wait_asynccnt)
#define USE_ASYNC_LDS 1
typedef __attribute__((address_space(1))) int GASInt;
typedef __attribute__((address_space(3))) int LASInt;
#else
#define USE_ASYNC_LDS 0
#endif

// Issue (without waiting) buf[c][tid] = logits[(bC_base + c)*CHW + hw], c=0..CC-1.
// Each lane issues CC async b32 loads (coalesced 1KB rows) tracked by ASYNCcnt.
__device__ __forceinline__ void issue_tile(const float* __restrict__ logits,
                                           int bC_base, int hw,
                                           float* __restrict__ buf) {
  const int tid = threadIdx.x;
#if USE_ASYNC_LDS
#pragma unroll
  for (int c = 0; c < CC; ++c) {
    const float* g = logits + (size_t)(bC_base + c) * (size_t)CHW + hw;
    __builtin_amdgcn_global_load_async_to_lds_b32(
        (GASInt*)g, (LASInt*)&buf[c * TPB + tid], 0, 0);
  }
#else
#pragma unroll
  for (int c = 0; c < CC; ++c) {
    buf[c * TPB + tid] = logits[(size_t)(bC_base + c) * (size_t)CHW + hw];
  }
#endif
}

// Wait so that only `remain` async ops are still in flight (in-order counter:
// the oldest ops — the previous tile — are then guaranteed complete).
__device__ __forceinline__ void wait_async_keep() {  // keep CC in flight
#if USE_ASYNC_LDS
  __builtin_amdgcn_s_wait_asynccnt(CC);
#endif
}
__device__ __forceinline__ void wait_async_all() {   // drain
#if USE_ASYNC_LDS
  __builtin_amdgcn_s_wait_asynccnt(0);
#endif
}

// Softmax over the thread's own tile column; caches exp(x-m) back in place
// (column `tid` is exclusively owned by this thread) and returns 1/sum.
__device__ __forceinline__ float softmax_col(float* __restrict__ col) {
  float m = col[0];
#pragma unroll
  for (int c = 1; c < CC; ++c) m = fmaxf(m, col[c * TPB]);
  float s = 0.f;
#pragma unroll
  for (int c = 0; c < CC; ++c) {
    const float e = __expf(col[c * TPB] - m);
    col[c * TPB] = e;
    s += e;
  }
  return 1.0f / s;
}

// ---------------- K0: zero workspace accumulators ----------------
__global__ void k_zero(unsigned int* __restrict__ w, int nwords) {
  int i = blockIdx.x * blockDim.x + threadIdx.x;
  if (i < nwords) w[i] = 0u;
}

// ---------------- K1: softmax + per-class fine histogram ----------------
// Bin mapping: bin = floor(sqrt(p) * HBINS)  (more resolution near p ~ 0,
// where most off-class confidences live). Inverse: p = ((bin+frac)/HBINS)^2.
__global__ __launch_bounds__(TPB) void k_hist(const float* __restrict__ logits,
                                              unsigned int* __restrict__ ghist) {
  __shared__ float tile[2][CC * TPB];        // 38.0 KB (double buffer)
  __shared__ unsigned int lhist[CC * HBINS]; // 38.9 KB
  const int tid = threadIdx.x;
  for (int i = tid; i < CC * HBINS; i += TPB) lhist[i] = 0u;

  const int n0 = blockIdx.x * PIX_PER_BLK;  // block-contiguous pixel chunk
  const int b = n0 / CHW;                   // constant over the chunk
  const int bC = b * CC;
  const int hw0 = n0 - b * CHW;

  issue_tile(logits, bC, hw0 + tid, tile[0]);  // prologue: tile 0 in flight
  __syncthreads();                             // covers lhist init too

  for (int t = 0; t < TILES; ++t) {
    const int cur = t & 1;
    if (t + 1 < TILES) {
      // other buffer was fully consumed in iter t-1 (barrier below protects it)
      issue_tile(logits, bC, hw0 + (t + 1) * TPB + tid, tile[cur ^ 1]);
      wait_async_keep();   // oldest CC ops (tile t) complete; tile t+1 streams on
    } else {
      wait_async_all();
    }
    __syncthreads();       // all waves' async writes to tile[cur] visible

    float* col = &tile[cur][tid];
    const float inv = softmax_col(col);
#pragma unroll
    for (int c = 0; c < CC; ++c) {
      const float p = col[c * TPB] * inv;
      int bin = (int)(sqrtf(p) * (float)HBINS);
      bin = bin < 0 ? 0 : (bin >= HBINS ? HBINS - 1 : bin);
      atomicAdd(&lhist[c * HBINS + bin], 1u);
    }
    __syncthreads();       // no wave may still read tile[cur^1]... / tile[cur] next overwrite
  }

  for (int i = tid; i < CC * HBINS; i += TPB) {
    unsigned int v = lhist[i];
    if (v) atomicAdd(&ghist[i], v);
  }
}

// ---------------- K2: inverse-CDF quantile edges per class ----------------
__global__ void k_edges(const unsigned int* __restrict__ ghist,
                        float* __restrict__ gedges) {
  const int c = threadIdx.x;
  if (c >= CC) return;
  const unsigned int* h = ghist + c * HBINS;
  float* e = gedges + c * NEDGES;
  const float step = (float)CN / (float)NBINS;  // target count per bin
  float cum = 0.f;
  int j = 0;
  for (int k = 0; k < HBINS && j <= NBINS; ++k) {
    const float cnt = (float)h[k];
    if (cnt > 0.f) {
      float tgt = step * (float)j;
      while (j <= NBINS && tgt <= cum + cnt) {
        float frac = (tgt - cum) / cnt;
        frac = frac < 0.f ? 0.f : (frac > 1.f ? 1.f : frac);
        const float u = ((float)k + frac) / (float)HBINS;
        e[j] = u * u;  // invert sqrt bin mapping
        ++j;
        tgt = step * (float)j;
      }
    }
    cum += cnt;
  }
  while (j <= NBINS) { e[j] = 1.0f; ++j; }
}

// ---------------- K3: bin assignment + per-(class,bin) stats ----------------
__global__ __launch_bounds__(TPB) void k_bin(const float* __restrict__ logits,
                                             const int* __restrict__ labels,
                                             const float* __restrict__ gedges,
                                             float* __restrict__ gstats) {
  __shared__ float tile[2][CC * TPB];         // 38.0 KB (double buffer)
  __shared__ float ledges[CC * NEDGES];       // 1.2 KB
  __shared__ float lstats[CC * NBINS * 3];    // 3.4 KB: {count, sum_conf, sum_corr}
  const int tid = threadIdx.x;
  for (int i = tid; i < CC * NEDGES; i += TPB) ledges[i] = gedges[i];
  for (int i = tid; i < CC * NBINS * 3; i += TPB) lstats[i] = 0.f;

  const int n0 = blockIdx.x * PIX_PER_BLK;
  const int b = n0 / CHW;
  const int bC = b * CC;
  const int hw0 = n0 - b * CHW;

  issue_tile(logits, bC, hw0 + tid, tile[0]);
  __syncthreads();

  for (int t = 0; t < TILES; ++t) {
    const int cur = t & 1;
    if (t + 1 < TILES) {
      issue_tile(logits, bC, hw0 + (t + 1) * TPB + tid, tile[cur ^ 1]);
      wait_async_keep();
    } else {
      wait_async_all();
    }
    __syncthreads();

    const int lab = labels[n0 + t * TPB + tid];
    float* col = &tile[cur][tid];
    const float inv = softmax_col(col);

#pragma unroll
    for (int c = 0; c < CC; ++c) {
      const float p = col[c * TPB] * inv;
      const float* e = &ledges[c * NEDGES];
      // searchsorted(e, p, side='left') == #edges strictly < p; bin j = idx-1
      int idx = 0;
#pragma unroll
      for (int i = 0; i < NEDGES; ++i) idx += (e[i] < p) ? 1 : 0;
      const int j = idx - 1;
      if (j >= 0 && j < NBINS) {
        float* st = &lstats[(c * NBINS + j) * 3];
        atomicAdd(st + 0, 1.0f);
        atomicAdd(st + 1, p);
        atomicAdd(st + 2, (lab == c) ? 1.0f : 0.0f);
      }
    }
    __syncthreads();
  }

  for (int i = tid; i < CC * NBINS * 3; i += TPB) {
    const float v = lstats[i];
    if (v != 0.f) atomicAdd(&gstats[i], v);
  }
}

// ---------------- K4: per-class ECE + mean ----------------
__global__ void k_final(const float* __restrict__ gstats, float* __restrict__ out) {
  __shared__ float part[32];
  const int c = threadIdx.x;
  float v = 0.f;
  if (c < CC) {
    for (int j = 0; j < NBINS; ++j) {
      const float* st = gstats + (c * NBINS + j) * 3;
      const float cnt = st[0];
      if (cnt > 0.f) {
        const float d = fmaxf(cnt, 1.0f);
        const float acc = st[2] / d;
        const float avg = st[1] / d;
        v += fabsf(avg - acc) * (cnt / (float)CN);
      }
    }
    out[1 + c] = v;
  }
  part[c] = (c < CC) ? v : 0.f;
  __syncthreads();
  if (c == 0) {
    float sum = 0.f;
    for (int i = 0; i < CC; ++i) sum += part[i];
    out[0] = sum / (float)CC;
  }
}

// ---------------- launch ----------------
extern "C" void kernel_launch(void* const* d_in, const int* in_sizes, int n_in,
                              void* d_out, int out_size, void* d_ws, size_t ws_size,
                              hipStream_t stream) {
  const float* logits = (const float*)d_in[0];
  const int* labels = (const int*)d_in[1];
  float* out = (float*)d_out;

  // workspace layout: hist[C][HBINS] (u32) | edges[C][16] (f32) | stats[C][15][3] (f32)
  unsigned int* hist = (unsigned int*)d_ws;
  float* edges = (float*)((char*)d_ws + (size_t)CC * HBINS * sizeof(unsigned int));
  float* stats = edges + CC * NEDGES;

  const int nwords = CC * HBINS + CC * NEDGES + CC * NBINS * 3;
  k_zero<<<(nwords + 255) / 256, 256, 0, stream>>>((unsigned int*)d_ws, nwords);
  k_hist<<<NBLK, TPB, 0, stream>>>(logits, hist);
  k_edges<<<1, 32, 0, stream>>>(hist, edges);
  k_bin<<<NBLK, TPB, 0, stream>>>(logits, labels, edges, stats);
  k_final<<<1, 32, 0, stream>>>(stats, out);
}